// RandomHingeForest_69114613728656
// MI455X (gfx1250) — compile-verified
//
#include <hip/hip_runtime.h>
#include <cstdint>
#include <cstddef>

// ---------------------------------------------------------------------------
// Random hinge forest forward, tuned for MI455X (gfx1250, wave32):
//  * x-row tile staged in LDS via gfx1250 async DMA (global_load_async_to_lds
//    _b128, ASYNCcnt) so the data-dependent feature gather is a ds_load.
//  * Tree node data packed {threshold, ordinal} into 8B records in d_ws by a
//    pre-pass, so each tree level costs ONE dependent global_load_b64 instead
//    of two scattered b32 loads (tree data is L2-resident: 33.5MB << 192MB).
//  * 8 independent traversal chains per thread to hide the 10-level latency
//    chain; 256-thread blocks = 8 wave32 waves; stores coalesced on t.
// ---------------------------------------------------------------------------

constexpr int kC       = 512;            // features per sample
constexpr int kDepth   = 10;
constexpr int kLeaves  = 1 << kDepth;    // 1024
constexpr int kNodes   = kLeaves - 1;    // 1023
constexpr int kBTile   = 32;             // samples per block: 32*512*4 = 64 KB LDS
constexpr int kTTile   = 256;            // trees per block (== blockDim.x)
constexpr int kThreads = 256;

typedef int v4i __attribute__((vector_size(16)));
typedef __attribute__((address_space(1))) v4i* global_v4i_ptr;
typedef __attribute__((address_space(3))) v4i* lds_v4i_ptr;

__device__ __forceinline__ void async_copy_b128(const float* gp, float* lp) {
#if defined(__AMDGCN__) && __has_builtin(__builtin_amdgcn_global_load_async_to_lds_b128)
  __builtin_amdgcn_global_load_async_to_lds_b128(
      (global_v4i_ptr)gp, (lds_v4i_ptr)lp, /*offset=*/0, /*cpol=*/0);
#else
  *(float4*)lp = *(const float4*)gp;
#endif
}

__device__ __forceinline__ void wait_async_zero() {
#if defined(__AMDGCN__) && __has_builtin(__builtin_amdgcn_s_wait_asynccnt)
  __builtin_amdgcn_s_wait_asynccnt(0);
#elif defined(__AMDGCN__)
  asm volatile("s_wait_asynccnt 0x0" ::: "memory");
#endif
}

// Branch-free LDS staging: trip count is compile-time, fully unrolled into 16
// straight async-DMA issues per thread (no exec-mask loop scaffolding).
__device__ __forceinline__ void stage_x_tile(const float* __restrict__ x,
                                             float* lds_x, int b0, int B) {
  constexpr int chunksPerRow = kC / 4;                  // 128 float4 per row
  constexpr int iters = (kBTile * chunksPerRow) / kThreads;  // 16
#pragma unroll
  for (int k = 0; k < iters; ++k) {
    const int idx = k * kThreads + (int)threadIdx.x;
    const int row = idx >> 7;                           // idx / 128
    const int col = idx & (chunksPerRow - 1);
    const int b   = min(b0 + row, B - 1);               // branchless clamp
    async_copy_b128(x + (size_t)b * kC + (size_t)col * 4,
                    &lds_x[row * kC + col * 4]);
  }
  wait_async_zero();
  __syncthreads();
}

// ---------------- pack pass: interleave {thr, ord} into 8B records ----------
__global__ __launch_bounds__(kThreads)
void pack_nodes(const float* __restrict__ thr, const int* __restrict__ ord,
                int2* __restrict__ packed, int n) {
  const int i = (int)(blockIdx.x * kThreads + threadIdx.x);
  if (i < n) {
    int2 rec;
    rec.x = __float_as_int(thr[i]);
    rec.y = ord[i];
    packed[i] = rec;                                    // coalesced b64 store
  }
}

// ---------------- main kernel, packed-node variant ---------------------------
template <int UNROLL>
__global__ __launch_bounds__(kThreads)
void hinge_forest_fwd_packed(const float* __restrict__ x,
                             const int2*  __restrict__ nodes,   // [T][1023]
                             const float* __restrict__ weights, // [T][1024]
                             float* __restrict__ out, int B, int T) {
  __shared__ float lds_x[kBTile * kC];   // 64 KB

  const int b0 = blockIdx.y * kBTile;
  const int t  = blockIdx.x * kTTile + (int)threadIdx.x;

  stage_x_tile(x, lds_x, b0, B);
  if (t >= T) return;

  const int2*  __restrict__ nd_t = nodes   + (size_t)t * kNodes;
  const float* __restrict__ wt_t = weights + (size_t)t * kLeaves;
  __builtin_prefetch(nd_t, 0, 3);        // warm top levels (global_prefetch_b8)

  for (int bb = 0; bb < kBTile; bb += UNROLL) {
    int   node[UNROLL];
    float minm[UNROLL];
#pragma unroll
    for (int u = 0; u < UNROLL; ++u) { node[u] = 0; minm[u] = __builtin_inff(); }

#pragma unroll
    for (int d = 0; d < kDepth; ++d) {
#pragma unroll
      for (int u = 0; u < UNROLL; ++u) {
        const int  n    = node[u];
        const int2 rec  = nd_t[n];                      // ONE b64 load / level
        const float thr = __int_as_float(rec.x);
        const float feat = lds_x[(bb + u) * kC + rec.y]; // LDS gather
        const float m    = feat - thr;
        const bool  take = __builtin_fabsf(m) < __builtin_fabsf(minm[u]);
        minm[u] = take ? m : minm[u];
        node[u] = 2 * n + 1 + (m > 0.0f ? 1 : 0);
      }
    }
#pragma unroll
    for (int u = 0; u < UNROLL; ++u) {
      const int b = b0 + bb + u;
      if (b < B)
        out[(size_t)b * T + t] = __builtin_fabsf(minm[u]) * wt_t[node[u] - kNodes];
    }
  }
}

// ---------------- fallback (no scratch): separate thr/ord loads --------------
__global__ __launch_bounds__(kThreads)
void hinge_forest_fwd(const float* __restrict__ x,
                      const float* __restrict__ thresholds,
                      const int*   __restrict__ ordinals,
                      const float* __restrict__ weights,
                      float* __restrict__ out, int B, int T) {
  __shared__ float lds_x[kBTile * kC];

  const int b0 = blockIdx.y * kBTile;
  const int t  = blockIdx.x * kTTile + (int)threadIdx.x;

  stage_x_tile(x, lds_x, b0, B);
  if (t >= T) return;

  const float* __restrict__ thr_t = thresholds + (size_t)t * kNodes;
  const int*   __restrict__ ord_t = ordinals   + (size_t)t * kNodes;
  const float* __restrict__ wt_t  = weights    + (size_t)t * kLeaves;
  __builtin_prefetch(thr_t, 0, 3);
  __builtin_prefetch(ord_t, 0, 3);

  constexpr int UNROLL = 4;
  for (int bb = 0; bb < kBTile; bb += UNROLL) {
    int   node[UNROLL];
    float minm[UNROLL];
#pragma unroll
    for (int u = 0; u < UNROLL; ++u) { node[u] = 0; minm[u] = __builtin_inff(); }
#pragma unroll
    for (int d = 0; d < kDepth; ++d) {
#pragma unroll
      for (int u = 0; u < UNROLL; ++u) {
        const int   n    = node[u];
        const float thr  = thr_t[n];
        const int   ord  = ord_t[n];
        const float feat = lds_x[(bb + u) * kC + ord];
        const float m    = feat - thr;
        const bool  take = __builtin_fabsf(m) < __builtin_fabsf(minm[u]);
        minm[u] = take ? m : minm[u];
        node[u] = 2 * n + 1 + (m > 0.0f ? 1 : 0);
      }
    }
#pragma unroll
    for (int u = 0; u < UNROLL; ++u) {
      const int b = b0 + bb + u;
      if (b < B)
        out[(size_t)b * T + t] = __builtin_fabsf(minm[u]) * wt_t[node[u] - kNodes];
    }
  }
}

extern "C" void kernel_launch(void* const* d_in, const int* in_sizes, int n_in,
                              void* d_out, int out_size, void* d_ws, size_t ws_size,
                              hipStream_t stream) {
  (void)n_in;
  const float* x          = (const float*)d_in[0];
  const float* thresholds = (const float*)d_in[1];
  const int*   ordinals   = (const int*)d_in[2];
  const float* weights    = (const float*)d_in[3];
  float*       out        = (float*)d_out;

  const int T = in_sizes[3] / kLeaves;   // weights is [T, 1024]
  const int B = out_size / T;            // out is [B, T]
  const int nNodes = T * kNodes;

  dim3 grid((T + kTTile - 1) / kTTile, (B + kBTile - 1) / kBTile);
  dim3 block(kThreads);

  const size_t packedBytes = (size_t)nNodes * sizeof(int2);
  if (d_ws != nullptr && ws_size >= packedBytes) {
    int2* packed = (int2*)d_ws;
    pack_nodes<<<(nNodes + kThreads - 1) / kThreads, block, 0, stream>>>(
        thresholds, ordinals, packed, nNodes);
    hinge_forest_fwd_packed<8><<<grid, block, 0, stream>>>(
        x, packed, weights, out, B, T);
  } else {
    hinge_forest_fwd<<<grid, block, 0, stream>>>(x, thresholds, ordinals,
                                                 weights, out, B, T);
  }
}